// DifferentiablePersistentHomology_90933047591278
// MI455X (gfx1250) — compile-verified
//
#include <hip/hip_runtime.h>
#include <math.h>

#define LM    50
#define NPTS  1024
#define TPB   256
#define PT    4          // points per thread = NPTS/TPB

typedef __attribute__((ext_vector_type(2))) float v2f;
typedef __attribute__((ext_vector_type(8))) float v8f;

// Monotone float->uint key (order-preserving) and inverse.
__device__ __forceinline__ unsigned fkey(float f) {
  unsigned b = __float_as_uint(f);
  return (b & 0x80000000u) ? ~b : (b | 0x80000000u);
}
__device__ __forceinline__ float funkey(unsigned u) {
  unsigned b = (u & 0x80000000u) ? (u ^ 0x80000000u) : ~u;
  return __uint_as_float(b);
}

struct Smem {
  float D[64 * 64];      // scaled distance matrix (64x64 padded)
  float Lmx[64], Lmy[64], norms[64];   // rows 50..63 stay zero (pad)
  float R[64];           // row sums
  int   hist[256];
  float redf[256];
  int   redi[256];
  int   scan[256];
  int   misc[4];
};

__device__ __forceinline__ float bsumf(float v, float* red) {
  int tid = threadIdx.x;
  red[tid] = v; __syncthreads();
  for (int s = TPB / 2; s > 0; s >>= 1) {
    if (tid < s) red[tid] += red[tid + s];
    __syncthreads();
  }
  float r = red[0]; __syncthreads();
  return r;
}
__device__ __forceinline__ float bminf(float v, float* red) {
  int tid = threadIdx.x;
  red[tid] = v; __syncthreads();
  for (int s = TPB / 2; s > 0; s >>= 1) {
    if (tid < s) red[tid] = fminf(red[tid], red[tid + s]);
    __syncthreads();
  }
  float r = red[0]; __syncthreads();
  return r;
}
__device__ __forceinline__ float bmaxf(float v, float* red) {
  int tid = threadIdx.x;
  red[tid] = v; __syncthreads();
  for (int s = TPB / 2; s > 0; s >>= 1) {
    if (tid < s) red[tid] = fmaxf(red[tid], red[tid + s]);
    __syncthreads();
  }
  float r = red[0]; __syncthreads();
  return r;
}
__device__ __forceinline__ int bsumi(int v, int* red) {
  int tid = threadIdx.x;
  red[tid] = v; __syncthreads();
  for (int s = TPB / 2; s > 0; s >>= 1) {
    if (tid < s) red[tid] += red[tid + s];
    __syncthreads();
  }
  int r = red[0]; __syncthreads();
  return r;
}

__global__ __launch_bounds__(TPB)
void dph_kernel(const float* __restrict__ pc,
                const float* __restrict__ attn_w,
                const float* __restrict__ fwp,
                const float* __restrict__ dbp,
                float* __restrict__ out) {
  __shared__ Smem s;
  const int tid = threadIdx.x;
  const int row = blockIdx.x;

  const float aw0 = attn_w[0];
  const float aw1 = attn_w[1];
  const float fwa = fabsf(fwp[0]);
  const float bias = dbp[0];

  // ---------- Phase 1: coalesced load of 4 points/thread + score keys ----------
  const float4* p4 = reinterpret_cast<const float4*>(pc + (size_t)row * (NPTS * 2));
  float4 q0 = p4[tid * 2];
  float4 q1 = p4[tid * 2 + 1];
  float px[PT] = {q0.x, q0.z, q1.x, q1.z};
  float py[PT] = {q0.y, q0.w, q1.y, q1.w};
  unsigned uk[PT];
#pragma unroll
  for (int e = 0; e < PT; ++e)
    uk[e] = fkey(fmaf(px[e], aw0, py[e] * aw1));   // softmax is monotone: rank raw scores

  // zero landmark pad rows (barriers inside the next phase make this visible)
  if (tid < 64) { s.Lmx[tid] = 0.f; s.Lmy[tid] = 0.f; s.norms[tid] = 0.f; }

  // ---------- Phase 2: radix-select 50th-largest score key ----------
  unsigned prefix = 0;
  int kk = LM;                                     // 1-based k-th largest
  for (int shift = 24; shift >= 0; shift -= 8) {
    s.hist[tid] = 0;
    __syncthreads();
    unsigned hiMask = (shift == 24) ? 0u : (0xFFFFFFFFu << (shift + 8));
#pragma unroll
    for (int e = 0; e < PT; ++e)
      if ((uk[e] & hiMask) == prefix)
        atomicAdd(&s.hist[(uk[e] >> shift) & 255], 1);
    __syncthreads();
    if (tid == 0) {
      int cnt = 0, c = 255;
      for (int b = 255; b >= 0; --b) {
        int h = s.hist[b];
        if (cnt + h >= kk) { c = b; break; }
        cnt += h;
      }
      s.misc[0] = c;
      s.misc[1] = kk - cnt;
    }
    __syncthreads();
    prefix |= ((unsigned)s.misc[0]) << shift;
    kk = s.misc[1];
    __syncthreads();
  }
  const unsigned tkey = prefix;

  // ---------- Phase 3: deterministic index-ordered gather of top-50 ----------
  int gt = 0, eq = 0;
#pragma unroll
  for (int e = 0; e < PT; ++e) { gt += (uk[e] > tkey); eq += (uk[e] == tkey); }
  int val = (gt << 16) | eq;
  s.scan[tid] = val;
  __syncthreads();
  for (int off = 1; off < TPB; off <<= 1) {        // Hillis-Steele inclusive scan
    int add = (tid >= off) ? s.scan[tid - off] : 0;
    __syncthreads();
    s.scan[tid] += add;
    __syncthreads();
  }
  int incl = s.scan[tid];
  int excl = incl - val;
  int total = s.scan[TPB - 1];
  int totalGt = total >> 16;
  int need = LM - totalGt;                         // ties filled by lowest index
  int gpre = excl >> 16;
  int epre = excl & 0xFFFF;
#pragma unroll
  for (int e = 0; e < PT; ++e) {
    int pos = -1;
    if (uk[e] > tkey) pos = gpre++;
    else if (uk[e] == tkey) { if (epre < need) pos = totalGt + epre; epre++; }
    if (pos >= 0 && pos < LM) {
      s.Lmx[pos] = px[e];
      s.Lmy[pos] = py[e];
      s.norms[pos] = fmaf(px[e], px[e], py[e] * py[e]);
    }
  }
  __syncthreads();

  // ---------- Phase 4: Gram via V_WMMA_F32_16X16X4_F32, 2 tiles/wave ----------
  // A 16x4: lanes 0-15 hold K=0,1; lanes 16-31 hold K=2,3 (zero pad).
  // Hi lanes read pad row 63 (guaranteed zero) -> unconditional ds_loads, no EXEC churn.
  const int wave = tid >> 5;
  const int lane = tid & 31;
  const bool lo = (lane < 16);
#pragma unroll
  for (int t = 0; t < 2; ++t) {
    int tileIdx = wave * 2 + t;
    int tm = (tileIdx >> 2) * 16;
    int tn = (tileIdx & 3) * 16;
    int mi = lo ? (tm + (lane & 15)) : 63;
    int ni = lo ? (tn + (lane & 15)) : 63;
    v2f a, b;
    a.x = s.Lmx[mi];
    a.y = s.Lmy[mi];
    b.x = s.Lmx[ni];
    b.y = s.Lmy[ni];
    v8f acc = {};
    acc = __builtin_amdgcn_wmma_f32_16x16x4_f32(false, a, false, b,
                                                (short)0, acc, false, false);
    int rbase = lo ? 0 : 8;
    float nj = s.norms[tn + (lane & 15)];
#pragma unroll
    for (int r = 0; r < 8; ++r) {
      int i = tm + rbase + r;
      int j = tn + (lane & 15);
      float d2 = s.norms[i] + nj - 2.0f * acc[r];
      // zero-safe sqrt matching the reference; fast HW v_sqrt_f32 (~1 ulp)
      float d = (i == j || !(d2 > 0.f)) ? 0.f : __builtin_amdgcn_sqrtf(d2);
      s.D[i * 64 + j] = fmaf(d, fwa, bias);
    }
  }
  __syncthreads();

  // ---------- Phase 5: statistics over the 50x50 valid region ----------
  float lsum = 0.f, lsq = 0.f, lmn = INFINITY, lmx = -INFINITY;
  for (int idx = tid; idx < LM * LM; idx += TPB) {
    int i = idx / LM, j = idx - i * LM;
    float v = s.D[i * 64 + j];
    lsum += v;
    lsq = fmaf(v, v, lsq);
    lmn = fminf(lmn, v);
    lmx = fmaxf(lmx, v);
  }
  float tsum = bsumf(lsum, s.redf);
  float tsq  = bsumf(lsq,  s.redf);
  float tmn  = bminf(lmn,  s.redf);
  float tmx  = bmaxf(lmx,  s.redf);

  // exact lower-median: 1250-th smallest of 2500 via radix-select
  unsigned mpre = 0;
  int mk = ((LM * LM - 1) / 2) + 1;                // 1-based rank 1250
  for (int shift = 24; shift >= 0; shift -= 8) {
    s.hist[tid] = 0;
    __syncthreads();
    unsigned hiMask = (shift == 24) ? 0u : (0xFFFFFFFFu << (shift + 8));
    for (int idx = tid; idx < LM * LM; idx += TPB) {
      int i = idx / LM, j = idx - i * LM;
      unsigned u = fkey(s.D[i * 64 + j]);
      if ((u & hiMask) == mpre)
        atomicAdd(&s.hist[(u >> shift) & 255], 1);
    }
    __syncthreads();
    if (tid == 0) {
      int cnt = 0, c = 0;
      for (int b2 = 0; b2 < 256; ++b2) {
        int h = s.hist[b2];
        if (cnt + h >= mk) { c = b2; break; }
        cnt += h;
      }
      s.misc[0] = c;
      s.misc[1] = mk - cnt;
    }
    __syncthreads();
    mpre |= ((unsigned)s.misc[0]) << shift;
    mk = s.misc[1];
    __syncthreads();
  }
  float med = funkey(mpre);

  int lcnt = 0;
  for (int idx = tid; idx < LM * LM; idx += TPB) {
    int i = idx / LM, j = idx - i * LM;
    lcnt += (s.D[i * 64 + j] < med);
  }
  int tcnt = bsumi(lcnt, s.redi);

  if (tid < LM) {
    float rs = 0.f;
    for (int j = 0; j < LM; ++j) rs += s.D[tid * 64 + j];
    s.R[tid] = rs;
  }
  __syncthreads();

  if (tid == 0) {
    const float n = (float)(LM * LM);
    float mean = tsum / n;
    float var = (tsq - tsum * mean) / (n - 1.0f);          // ddof=1
    float rsum = 0.f, rsq = 0.f;
    for (int i = 0; i < LM; ++i) { float v = s.R[i]; rsum += v; rsq = fmaf(v, v, rsq); }
    float rmean = rsum / (float)LM;
    float rvar = (rsq - rsum * rmean) / (float)(LM - 1);
    float* o = out + (size_t)row * 6;
    o[0] = mean;
    o[1] = sqrtf(fmaxf(var, 0.f));
    o[2] = tmn;
    o[3] = tmx;
    o[4] = (float)tcnt / n;
    o[5] = sqrtf(fmaxf(rvar, 0.f));
  }
}

extern "C" void kernel_launch(void* const* d_in, const int* in_sizes, int n_in,
                              void* d_out, int out_size, void* d_ws, size_t ws_size,
                              hipStream_t stream) {
  const float* pc     = (const float*)d_in[0];  // (B, 1024, 2)
  const float* attn_w = (const float*)d_in[1];  // (1, 2)
  // d_in[2] = attn_b: constant shift, does not change top-k ordering -> unused
  const float* fw     = (const float*)d_in[3];  // (1, 1)
  const float* db     = (const float*)d_in[4];  // (1,)
  float* out = (float*)d_out;                   // (B, 6) float32

  int B = in_sizes[0] / (NPTS * 2);
  dph_kernel<<<dim3(B), dim3(TPB), 0, stream>>>(pc, attn_w, fw, db, out);
}